// GatherFromIndices_7902739825140
// MI455X (gfx1250) — compile-verified
//
#include <hip/hip_runtime.h>

// Problem constants from the reference (fixed by setup_inputs()).
constexpr int Bc  = 8;
constexpr int Nc  = 10000;
constexpr int Fc  = 64;      // floats per row -> 256 bytes
constexpr int Kc  = 8;
constexpr int KP1 = Kc + 1;  // 9 output slots per (b,n)
constexpr int SLOTS = Bc * Nc * KP1;   // 720,000 rows of 64 floats

constexpr int THREADS = 256;           // 8 wave32s
constexpr int UNROLL  = 4;             // slots per 16-lane group per block
constexpr int TILE    = (THREADS / 16) * UNROLL;   // 64 slots per block
constexpr int BLOCKS  = SLOTS / TILE;  // 11,250 exactly (no remainder)

// ---- gfx1250 async global<->LDS ops (ASYNCcnt path), via inline asm ----
// load: copies 16 bytes from the global address in vaddr to the LDS byte address in vdst
__device__ __forceinline__ void async_load_b128(unsigned lds_addr,
                                                const void* gaddr) {
    asm volatile("global_load_async_to_lds_b128 %0, %1, off"
                 :
                 : "v"(lds_addr), "v"(gaddr)
                 : "memory");
}
// store: copies 16 bytes from the LDS byte address in vsrc to the global address in vaddr
// NT temporal hint: output is 184 MB write-once -> keep gather table in L2.
__device__ __forceinline__ void async_store_b128(void* gaddr,
                                                 unsigned lds_addr) {
    asm volatile("global_store_async_from_lds_b128 %0, %1, off th:TH_STORE_NT"
                 :
                 : "v"(gaddr), "v"(lds_addr)
                 : "memory");
}
__device__ __forceinline__ void wait_asynccnt0() {
    asm volatile("s_wait_asynccnt 0x0" ::: "memory");
}

// One 16-lane group (half a wave32) owns one 256-byte row: lane i DMAs
// bytes [16i,16i+16) global->LDS, then LDS->global. UNROLL=4 rows per
// group are batched so each lane keeps 4 async transfers in flight on
// ASYNCcnt before the single s_wait_asynccnt 0. LDS regions are
// per-lane-private, so no workgroup barrier is required.
__global__ __launch_bounds__(THREADS) void gather_async_kernel(
    const float* __restrict__ inp,    // [B*N, F]
    const int*   __restrict__ inds,   // [B*N, K]
    const float* __restrict__ zrow,   // [F] zeros (sentinel row)
    float*       __restrict__ out)    // [B*N, K+1, F]
{
    __shared__ __align__(16) unsigned char smem[THREADS * 16 * UNROLL];

    const int lane16 = threadIdx.x & 15;   // float4 slot within the row
    const int group  = threadIdx.x >> 4;   // 16-lane group id (0..15)

    // ISA: flat LDS aperture keeps the LDS byte offset in addr[31:0],
    // so truncating the flat pointer yields the wave-relative LDS address.
    const unsigned lds0 =
        (unsigned)(uintptr_t)(&smem[0]) + (unsigned)threadIdx.x * 16u;

    const int tileBase = blockIdx.x * TILE;

    void* gdst[UNROLL];

#pragma unroll
    for (int j = 0; j < UNROLL; ++j) {
        const int slot = tileBase + j * 16 + group;
        // slot -> (bn, k) -> (b, n): constant divisors lower to
        // magic-number multiplies.
        const int k  = slot % KP1;
        const int bn = slot / KP1;          // b*N + n
        const int b  = bn / Nc;

        const float* src;
        if (k == Kc) {
            src = inp + (size_t)bn * Fc;    // self row
        } else {
            const int idx = inds[bn * Kc + k];
            // Faithful to reference: masked -> sentinel zero row,
            // else (idx % N) + b*N.
            src = (idx >= 0) ? (inp + (size_t)(b * Nc + (idx % Nc)) * Fc)
                             : zrow;
        }

        async_load_b128(lds0 + (unsigned)j * (THREADS * 16u),
                        (const void*)(src + lane16 * 4));
        gdst[j] = (void*)(out + (size_t)slot * Fc + lane16 * 4);
    }

    // All 4 gather DMAs into this lane's LDS regions must land before the
    // outbound DMAs read them.
    wait_asynccnt0();

#pragma unroll
    for (int j = 0; j < UNROLL; ++j) {
        async_store_b128(gdst[j], lds0 + (unsigned)j * (THREADS * 16u));
    }
    // S_ENDPGM performs an implicit wait-idle, draining the async stores.
}

// Prologue: build the sentinel MASK_VALUE row (mirrors the reference's
// appended zero row) in workspace. Re-run every call (d_ws is poisoned).
__global__ void zero_row_kernel(float* __restrict__ z) {
    z[threadIdx.x] = 0.0f;
}

extern "C" void kernel_launch(void* const* d_in, const int* in_sizes, int n_in,
                              void* d_out, int out_size, void* d_ws, size_t ws_size,
                              hipStream_t stream) {
    (void)in_sizes; (void)n_in; (void)out_size; (void)ws_size;

    const float* inp  = (const float*)d_in[0];  // [8,10000,64] f32
    const int*   inds = (const int*)d_in[1];    // [8,10000,8] i32
    float*       out  = (float*)d_out;          // [8,10000,9,64] f32
    float*       zrow = (float*)d_ws;           // 64 floats of scratch

    // Stream-ordered: sentinel row is ready before the gather kernel runs.
    zero_row_kernel<<<1, Fc, 0, stream>>>(zrow);
    gather_async_kernel<<<BLOCKS, THREADS, 0, stream>>>(inp, inds, zrow, out);
}